// SpatialAttnBias_12283606466663
// MI455X (gfx1250) — compile-verified
//
#include <hip/hip_runtime.h>

// ---------------------------------------------------------------------------
// SpatialAttnBias: blocked Floyd-Warshall (min-plus, f32) + embedding gather.
//   V = 1536, tile 64x64, 24 rounds of {phase1, phase2, phase3}, then gather.
// CDNA5 paths: global_load_async_to_lds_b128 (ASYNCcnt) for tile staging,
// s_wait_asynccnt, ds_load_b128 from LDS, wave32-sized 256-thread blocks.
// ---------------------------------------------------------------------------

#define VN    1536
#define TB    64
#define NBLK  (VN / TB)          // 24
#define PADW  68                 // LDS row stride in floats (272B, 16B aligned)
#define NSP   769
#define BD    16

#if defined(__has_builtin)
#  if __has_builtin(__builtin_amdgcn_global_load_async_to_lds_b128)
#    define HAVE_ASYNC_LDS 1
#  endif
#endif

#ifdef HAVE_ASYNC_LDS
// Matches the builtin's expected parameter types (from clang diagnostic):
// pointer to 16-byte int vector, addrspace(1) for global / addrspace(3) for LDS.
typedef int v4i __attribute__((vector_size(4 * sizeof(int))));
typedef __attribute__((address_space(1))) v4i* g_v4i_ptr;
typedef __attribute__((address_space(3))) v4i* l_v4i_ptr;
#endif

__device__ __forceinline__ void async_b128(const float* g, float* l) {
#ifdef HAVE_ASYNC_LDS
  __builtin_amdgcn_global_load_async_to_lds_b128((g_v4i_ptr)g, (l_v4i_ptr)l, 0, 0);
#else
  *(float4*)l = *(const float4*)g;
#endif
}

__device__ __forceinline__ void wait_async_lds() {
#ifdef HAVE_ASYNC_LDS
#  if __has_builtin(__builtin_amdgcn_s_wait_asynccnt)
  __builtin_amdgcn_s_wait_asynccnt(0);
#  else
  asm volatile("s_wait_asynccnt 0" ::: "memory");
#  endif
#endif
}

// Load a TB x TB tile (row-major, row stride VN) into LDS (row stride PADW).
// 256 threads: each thread copies 16 consecutive floats of one row (4x b128).
__device__ __forceinline__ void load_tile_async(const float* gbase, float* lds) {
  const int t     = threadIdx.x;
  const int row   = t >> 2;
  const int chunk = (t & 3) * 16;
  const float* g = gbase + (size_t)row * VN + chunk;
  float*       l = lds + row * PADW + chunk;
  async_b128(g + 0,  l + 0);
  async_b128(g + 4,  l + 4);
  async_b128(g + 8,  l + 8);
  async_b128(g + 12, l + 12);
}

__device__ __forceinline__ void store_tile(float* gbase, const float* lds) {
  const int t     = threadIdx.x;
  const int row   = t >> 2;
  const int chunk = (t & 3) * 16;
  float*       g = gbase + (size_t)row * VN + chunk;
  const float* l = lds + row * PADW + chunk;
  *(float4*)(g + 0)  = *(const float4*)(l + 0);
  *(float4*)(g + 4)  = *(const float4*)(l + 4);
  *(float4*)(g + 8)  = *(const float4*)(l + 8);
  *(float4*)(g + 12) = *(const float4*)(l + 12);
}

__device__ __forceinline__ float4 mp_upd(float4 c, float a, float4 b) {
  c.x = fminf(c.x, a + b.x);
  c.y = fminf(c.y, a + b.y);
  c.z = fminf(c.z, a + b.z);
  c.w = fminf(c.w, a + b.w);
  return c;
}

// d[i*V+j] = (g==0 && i!=j) ? +inf : g   (faithful to reference _modify_graph)
__global__ __launch_bounds__(256) void fw_init(const float* __restrict__ g,
                                               float* __restrict__ d) {
  const int p = blockIdx.x * 256 + threadIdx.x;
  const int i = p / VN;
  const int j = p - i * VN;
  const float w = g[p];
  d[p] = (w == 0.0f && i != j) ? __builtin_inff() : w;
}

// Phase 1: FW inside the pivot diagonal tile. One block.
// Races at pivot row/col are value-preserving because C[k][k] == 0.
__global__ __launch_bounds__(256) void fw_phase1(float* __restrict__ d, int r) {
  __shared__ float C[TB * PADW];
  float* gbase = d + (size_t)(r * TB) * VN + r * TB;
  load_tile_async(gbase, C);
  wait_async_lds();
  __syncthreads();

  const int tx = threadIdx.x & 15, ty = threadIdx.x >> 4;
  const int i0 = ty * 4, j0 = tx * 4;
  for (int k = 0; k < TB; ++k) {
    float4 bk = *(const float4*)&C[k * PADW + j0];
#pragma unroll
    for (int rr = 0; rr < 4; ++rr) {
      const float a = C[(i0 + rr) * PADW + k];
      float4 cv = *(float4*)&C[(i0 + rr) * PADW + j0];
      cv = mp_upd(cv, a, bk);
      *(float4*)&C[(i0 + rr) * PADW + j0] = cv;
    }
    __syncthreads();
  }
  store_tile(gbase, C);
}

// Phase 2: pivot row blocks (y==0) and pivot column blocks (y==1).
__global__ __launch_bounds__(256) void fw_phase2(float* __restrict__ d, int r) {
  const int x = blockIdx.x;
  if (x == r) return;
  const int isCol = blockIdx.y;

  __shared__ float C[TB * PADW];
  __shared__ float D[TB * PADW];
  float* cbase = isCol ? d + (size_t)(x * TB) * VN + r * TB
                       : d + (size_t)(r * TB) * VN + x * TB;
  float* dbase = d + (size_t)(r * TB) * VN + r * TB;
  load_tile_async(cbase, C);
  load_tile_async(dbase, D);
  wait_async_lds();
  __syncthreads();

  const int tx = threadIdx.x & 15, ty = threadIdx.x >> 4;
  const int i0 = ty * 4, j0 = tx * 4;
  if (!isCol) {
    // C[i][j] = min(C[i][j], D[i][k] + C[k][j])
    for (int k = 0; k < TB; ++k) {
      float4 bk = *(const float4*)&C[k * PADW + j0];
#pragma unroll
      for (int rr = 0; rr < 4; ++rr) {
        const float a = D[(i0 + rr) * PADW + k];
        float4 cv = *(float4*)&C[(i0 + rr) * PADW + j0];
        cv = mp_upd(cv, a, bk);
        *(float4*)&C[(i0 + rr) * PADW + j0] = cv;
      }
      __syncthreads();
    }
  } else {
    // C[i][j] = min(C[i][j], C[i][k] + D[k][j])
    for (int k = 0; k < TB; ++k) {
      float4 bk = *(const float4*)&D[k * PADW + j0];
#pragma unroll
      for (int rr = 0; rr < 4; ++rr) {
        const float a = C[(i0 + rr) * PADW + k];
        float4 cv = *(float4*)&C[(i0 + rr) * PADW + j0];
        cv = mp_upd(cv, a, bk);
        *(float4*)&C[(i0 + rr) * PADW + j0] = cv;
      }
      __syncthreads();
    }
  }
  store_tile(cbase, C);
}

// Phase 3: min-plus GEMM  C(bi,bj) = min(C, A(bi,r) (+) B(r,bj)).
// No intra-block dependency: A,B are final for this round. 4x4 regs/thread.
__global__ __launch_bounds__(256) void fw_phase3(float* __restrict__ d, int r) {
  const int bj = blockIdx.x, bi = blockIdx.y;
  if (bi == r || bj == r) return;

  __shared__ float At[TB * TB];    // transposed A: At[k*TB + i]
  __shared__ float Bs[TB * PADW];  // row-major B
  const float* abase = d + (size_t)(bi * TB) * VN + r * TB;
  const float* bbase = d + (size_t)(r * TB) * VN + bj * TB;
  float*       cbase = d + (size_t)(bi * TB) * VN + bj * TB;

  load_tile_async(bbase, Bs);  // async engine fills Bs while we transpose A

  {
    const int t = threadIdx.x;
    const int row = t >> 2;
    const int chunk = (t & 3) * 16;
    const float* g = abase + (size_t)row * VN + chunk;
#pragma unroll
    for (int q = 0; q < 4; ++q) {
      const float4 v = *(const float4*)(g + q * 4);
      const int c0i = chunk + q * 4;
      At[(c0i + 0) * TB + row] = v.x;
      At[(c0i + 1) * TB + row] = v.y;
      At[(c0i + 2) * TB + row] = v.z;
      At[(c0i + 3) * TB + row] = v.w;
    }
  }

  const int tx = threadIdx.x & 15, ty = threadIdx.x >> 4;
  const int i0 = ty * 4, j0 = tx * 4;
  float* crow = cbase + (size_t)i0 * VN + j0;
  float4 c0 = *(float4*)(crow);
  float4 c1 = *(float4*)(crow + VN);
  float4 c2 = *(float4*)(crow + 2 * VN);
  float4 c3 = *(float4*)(crow + 3 * VN);

  wait_async_lds();
  __syncthreads();

#pragma unroll 4
  for (int k = 0; k < TB; ++k) {
    const float4 a = *(const float4*)&At[k * TB + i0];
    const float4 b = *(const float4*)&Bs[k * PADW + j0];
    c0 = mp_upd(c0, a.x, b);
    c1 = mp_upd(c1, a.y, b);
    c2 = mp_upd(c2, a.z, b);
    c3 = mp_upd(c3, a.w, b);
  }

  *(float4*)(crow)          = c0;
  *(float4*)(crow + VN)     = c1;
  *(float4*)(crow + 2 * VN) = c2;
  *(float4*)(crow + 3 * VN) = c3;
}

// Gather: out[p, 0:16] = table[(int)d[p], 0:16], table staged in LDS (49 KB).
__global__ __launch_bounds__(256) void gather_bias(const float* __restrict__ d,
                                                   const float* __restrict__ table,
                                                   float* __restrict__ out) {
  __shared__ float tb[NSP * BD];
  for (int i = threadIdx.x; i < NSP * BD; i += 256) tb[i] = table[i];
  __syncthreads();

  const int total  = VN * VN;
  const int stride = gridDim.x * 256;
  for (int p = blockIdx.x * 256 + threadIdx.x; p < total; p += stride) {
    const int si = (int)d[p];
    const float4* row = (const float4*)&tb[si * BD];
    float4* o = (float4*)(out + (size_t)p * BD);
    o[0] = row[0];
    o[1] = row[1];
    o[2] = row[2];
    o[3] = row[3];
  }
}

extern "C" void kernel_launch(void* const* d_in, const int* in_sizes, int n_in,
                              void* d_out, int out_size, void* d_ws, size_t ws_size,
                              hipStream_t stream) {
  (void)in_sizes; (void)n_in; (void)out_size; (void)ws_size;
  const float* graph = (const float*)d_in[0];   // [V,V] f32
  const float* table = (const float*)d_in[1];   // [769,16] f32
  float* out = (float*)d_out;                   // [V,V,16] f32
  float* dmat = (float*)d_ws;                   // [V,V] f32 scratch (9.4 MB)

  fw_init<<<(VN * VN) / 256, 256, 0, stream>>>(graph, dmat);
  for (int r = 0; r < NBLK; ++r) {
    fw_phase1<<<1, 256, 0, stream>>>(dmat, r);
    fw_phase2<<<dim3(NBLK, 2), 256, 0, stream>>>(dmat, r);
    fw_phase3<<<dim3(NBLK, NBLK), 256, 0, stream>>>(dmat, r);
  }
  gather_bias<<<1152, 256, 0, stream>>>(dmat, table, out);
}